// Model_26620207301097
// MI455X (gfx1250) — compile-verified
//
#include <hip/hip_runtime.h>

// Embedding gather: out[i, 0:16] = table[indices[i], 0:16]
//
// MI455X strategy (pure bandwidth problem, ~350 MB HBM traffic, ~15 us floor
// at 23.3 TB/s):
//  - 64 MB table is L2-resident (192 MB L2): table loads use default
//    (temporal) policy so they stay cached.
//  - 256 MiB output stream and 16 MiB index stream use NON-TEMPORAL
//    hints so they do not evict the table from L2.
//  - 4 lanes cooperate per 64-byte row, each moving one float4 (b128 ops);
//    a wave32 store instruction covers a contiguous 512 B span.

typedef float f32x4 __attribute__((ext_vector_type(4)));

__global__ __launch_bounds__(256) void embed_gather_kernel(
    const int* __restrict__ indices,
    const f32x4* __restrict__ table,   // viewed as [VOCAB*4] float4 chunks
    f32x4* __restrict__ out,           // viewed as [NROWS*4] float4 chunks
    int nrows) {
  const int t   = blockIdx.x * blockDim.x + threadIdx.x;
  const int row = t >> 2;      // output row handled by this lane-group of 4
  const int sub = t & 3;       // which 16-byte chunk of the 64-byte row

  if (row < nrows) {
    // Index line is consumed entirely by this wave in one instruction:
    // non-temporal load so it is evicted first, protecting the table in L2.
    const int r = __builtin_nontemporal_load(&indices[row]);

    // Table gather: default (temporal) policy -> 64 MB table stays resident
    // in the 192 MB L2, so random row reads hit L2 after first touch.
    const f32x4 v = table[(size_t)r * 4 + (size_t)sub];

    // Output is write-once streamed: non-temporal b128 store
    // (global_store_b128 th:TH_STORE_NT) rinses through without polluting L2.
    __builtin_nontemporal_store(v, &out[(size_t)row * 4 + (size_t)sub]);
  }
}

extern "C" void kernel_launch(void* const* d_in, const int* in_sizes, int n_in,
                              void* d_out, int out_size, void* d_ws, size_t ws_size,
                              hipStream_t stream) {
  const int*   indices = (const int*)d_in[0];     // int32 indices
  const float* table   = (const float*)d_in[1];   // [VOCAB, 16] f32
  float*       out     = (float*)d_out;           // [NROWS, 16] f32

  const int nrows   = in_sizes[0];                // 4,194,304
  const int threads = nrows * 4;                  // 4 lanes per row
  const int block   = 256;                        // 8 wave32s per block
  const int grid    = (threads + block - 1) / block;

  embed_gather_kernel<<<grid, block, 0, stream>>>(
      indices, (const f32x4*)table, (f32x4*)out, nrows);
}